// Yolov3Post_80358838108772
// MI455X (gfx1250) — compile-verified
//
#include <hip/hip_runtime.h>
#include <cstdint>

#define NUM_CLASSES 80
#define CONF_TH     0.25f
#define IOU_TH      0.45f
#define MAX_DET     100
#define NIMG        16
#define NBOX        22743              // 3*(76*76 + 38*38 + 19*19)
#define NPAD        22752              // NBOX rounded up to multiple of 32 (and 4)
#define OFF_S8      0
#define OFF_S16     17328              // 3*5776
#define OFF_S32     21660              // 3*5776 + 3*1444
#define NMS_THREADS 1024
#define NMS_WAVES   (NMS_THREADS / 32)

typedef uint32_t u32x4 __attribute__((ext_vector_type(4)));
typedef int32_t  i32x8 __attribute__((ext_vector_type(8)));
typedef int32_t  i32x4 __attribute__((ext_vector_type(4)));

// ---------------------------------------------------------------------------
// Decode kernel: one thread per (b, anchor, y, x) box. H/W are template
// constants so all index math strength-reduces (no runtime div/mod chains).
// All 85 channel loads are wave-coalesced (x fastest, channels strided by
// H*W) and non-temporal: features are streamed once, keep L2 for boxes/scores.
// Sigmoid is monotone => argmax over class sigmoids == argmax over logits,
// so only ONE sigmoid of the max logit is needed (85 loads, ~80 max, 6 exp).
// ---------------------------------------------------------------------------
template <int H, int W>
__global__ __launch_bounds__(256)
void yolo_decode_kernel(const float* __restrict__ feat,
                        float4* __restrict__ boxes,
                        float*  __restrict__ scores,
                        int*    __restrict__ classes,
                        float stride,
                        float aw0, float ah0, float aw1, float ah1,
                        float aw2, float ah2,
                        int level_off)
{
    constexpr int HW = H * W;
    constexpr int total = NIMG * 3 * HW;
    int t = blockIdx.x * blockDim.x + threadIdx.x;
    if (t >= total) return;

    const int x = t % W;
    const int y = (t / W) % H;
    const int a = (t / HW) % 3;
    const int b = t / (3 * HW);

    const float* f = feat + (size_t)(b * 255 + a * 85) * HW + (size_t)y * W + x;

    const float tx = __builtin_nontemporal_load(f);
    const float ty = __builtin_nontemporal_load(f + (size_t)1 * HW);
    const float tw = __builtin_nontemporal_load(f + (size_t)2 * HW);
    const float th = __builtin_nontemporal_load(f + (size_t)3 * HW);
    const float to = __builtin_nontemporal_load(f + (size_t)4 * HW);

    // max/argmax over the 80 class logits (strict '>' keeps first occurrence,
    // matching jnp.argmax semantics)
    float m  = __builtin_nontemporal_load(f + (size_t)5 * HW);
    int   ci = 0;
#pragma unroll 4
    for (int c = 1; c < NUM_CLASSES; ++c) {
        float v = __builtin_nontemporal_load(f + (size_t)(5 + c) * HW);
        if (v > m) { m = v; ci = c; }
    }

    const float sx = 1.f / (1.f + expf(-tx));
    const float sy = 1.f / (1.f + expf(-ty));
    const float so = 1.f / (1.f + expf(-to));
    const float sc = 1.f / (1.f + expf(-m));

    const float aw = (a == 0) ? aw0 : (a == 1) ? aw1 : aw2;
    const float ah = (a == 0) ? ah0 : (a == 1) ? ah1 : ah2;

    const float cx = (sx + (float)x) * stride;
    const float cy = (sy + (float)y) * stride;
    const float w  = expf(tw) * aw;
    const float h  = expf(th) * ah;

    float score = so * sc;
    score = (score > CONF_TH) ? score : 0.f;

    const int n = b * NPAD + level_off + (a * H + y) * W + x;
    boxes[n]   = make_float4(cx - 0.5f * w, cy - 0.5f * h,
                             cx + 0.5f * w, cy + 0.5f * h);
    scores[n]  = score;
    classes[n] = ci;
}

// packed argmax key: non-negative score bits (order-preserving) in the high
// word, (NPAD - index) in the low word so max() tie-breaks to min index
// (== jnp.argmax first-occurrence). key==0 is below every real candidate.
__device__ __forceinline__ uint64_t amax_key(float v, int i) {
    return ((uint64_t)__float_as_uint(v) << 32) | (uint32_t)(NPAD - i);
}

// ---------------------------------------------------------------------------
// NMS kernel: one workgroup (32 waves) per image. The 22743-entry score
// vector (91 KB) is DMA'd global->LDS with the CDNA5 Tensor Data Mover
// (tensor_load_to_lds / TENSORcnt); the 100 greedy rounds then run entirely
// out of LDS (b128 scan, wave32 u64 shuffle reductions). Boxes (5.8 MB total)
// stay L2-resident and are read only for still-live candidates.
// ---------------------------------------------------------------------------
__global__ __launch_bounds__(NMS_THREADS)
void yolo_nms_kernel(const float4* __restrict__ boxes,
                     const float*  __restrict__ scores_g,
                     const int*    __restrict__ classes,
                     float* __restrict__ out)
{
    extern __shared__ float smem[];
    float*    s_sc   = smem;                               // [NPAD]
    uint64_t* s_key  = (uint64_t*)(smem + NPAD);           // [NMS_WAVES]
    float*    s_best = (float*)(s_key + NMS_WAVES);        // [5]

    const int b   = blockIdx.x;
    const int tid = threadIdx.x;

    // --- Stage this image's scores into LDS via the Tensor Data Mover -----
    if (tid < 32) {   // TDM ignores EXEC: issue from exactly one wave
        const uint32_t lds_off = (uint32_t)(uintptr_t)s_sc;   // LDS byte addr
        const uint64_t gaddr =
            (uint64_t)(uintptr_t)(scores_g + (size_t)b * NPAD);

        // D# group 0: count=1 (valid user desc), lds_addr, global_addr, type=2
        u32x4 g0;
        g0[0] = 1u;
        g0[1] = lds_off;
        g0[2] = (uint32_t)gaddr;
        g0[3] = ((uint32_t)(gaddr >> 32) & 0x1FFFFFFu) | (2u << 30);

        // D# group 1: data_size=4B; tensor = NBOX x 1; tile = NBOX x 1
        i32x8 g1;
        g1[0] = (int)(2u << 16);                 // data_size=2 -> 4 bytes
        g1[1] = (int)((uint32_t)NBOX << 16);     // tensor_dim0[15:0]
        g1[2] = (int)(1u << 16);                 // tensor_dim0 hi=0, dim1=1
        g1[3] = (int)((uint32_t)NBOX << 16);     // tile_dim0 = NBOX
        g1[4] = 1;                               // tile_dim1 = 1
        g1[5] = NPAD;                            // tensor_dim0_stride
        g1[6] = 0;
        g1[7] = 0;

        i32x4 g2 = {0, 0, 0, 0};
        i32x4 g3 = {0, 0, 0, 0};

        asm volatile("tensor_load_to_lds %0, %1, %2, %3"
                     :: "s"(g0), "s"(g1), "s"(g2), "s"(g3)
                     : "memory");
        __builtin_amdgcn_s_wait_tensorcnt(0);
    }
    // zero the pad tail so argmax/suppression never see junk
    for (int i = NBOX + tid; i < NPAD; i += NMS_THREADS) s_sc[i] = 0.f;
    __syncthreads();

    for (int it = 0; it < MAX_DET; ++it) {
        // ---- block-wide packed argmax, b128 LDS scan ---------------------
        const float4* s_sc4 = (const float4*)s_sc;
        uint64_t key = 0;
        for (int i4 = tid; i4 < NPAD / 4; i4 += NMS_THREADS) {
            float4 q = s_sc4[i4];
            const int base = i4 * 4;
            uint64_t k;
            k = amax_key(q.x, base);     if (k > key) key = k;
            k = amax_key(q.y, base + 1); if (k > key) key = k;
            k = amax_key(q.z, base + 2); if (k > key) key = k;
            k = amax_key(q.w, base + 3); if (k > key) key = k;
        }
        for (int off = 16; off > 0; off >>= 1) {
            uint64_t o = __shfl_down(key, off, 32);
            if (o > key) key = o;
        }
        if ((tid & 31) == 0) s_key[tid >> 5] = key;
        __syncthreads();

        if (tid == 0) {
            uint64_t kk = s_key[0];
            for (int wv = 1; wv < NMS_WAVES; ++wv)
                if (s_key[wv] > kk) kk = s_key[wv];
            const int   bi = NPAD - (int)(kk & 0xFFFFFFFFu);
            const float bv = __uint_as_float((uint32_t)(kk >> 32));
            const float4 box = boxes[(size_t)b * NPAD + bi];
            if (bv > 0.f) {
                float* o = out + ((size_t)b * MAX_DET + it) * 6;
                o[0] = box.x; o[1] = box.y; o[2] = box.z; o[3] = box.w;
                o[4] = bv;
                o[5] = (float)classes[(size_t)b * NPAD + bi];
                s_sc[bi] = 0.f;
            }
            s_best[0] = box.x; s_best[1] = box.y;
            s_best[2] = box.z; s_best[3] = box.w;
            s_best[4] = bv;
        }
        __syncthreads();

        const float bv = s_best[4];
        if (bv <= 0.f) {
            // everything left is a pad row: fill collectively and stop
            for (int e = it * 6 + tid; e < MAX_DET * 6; e += NMS_THREADS)
                out[(size_t)b * MAX_DET * 6 + e] = ((e % 6) == 5) ? -1.f : 0.f;
            break;
        }

        const float bx1 = s_best[0], by1 = s_best[1];
        const float bx2 = s_best[2], by2 = s_best[3];
        const float a1  = (bx2 - bx1) * (by2 - by1);
        for (int i = tid; i < NPAD; i += NMS_THREADS) {
            float v = s_sc[i];
            if (v > 0.f) {
                float4 c = boxes[(size_t)b * NPAD + i];
                float lx = fmaxf(bx1, c.x), ly = fmaxf(by1, c.y);
                float rx = fminf(bx2, c.z), ry = fminf(by2, c.w);
                float iw = fmaxf(rx - lx, 0.f);
                float ih = fmaxf(ry - ly, 0.f);
                float inter = iw * ih;
                float a2 = (c.z - c.x) * (c.w - c.y);
                float iou = inter / (a1 + a2 - inter + 1e-9f);
                if (iou > IOU_TH) s_sc[i] = 0.f;
            }
        }
        __syncthreads();
    }
}

// ---------------------------------------------------------------------------
extern "C" void kernel_launch(void* const* d_in, const int* in_sizes, int n_in,
                              void* d_out, int out_size, void* d_ws, size_t ws_size,
                              hipStream_t stream)
{
    (void)in_sizes; (void)n_in; (void)out_size; (void)ws_size;
    const float* f8  = (const float*)d_in[0];   // (16,255,76,76)
    const float* f16 = (const float*)d_in[1];   // (16,255,38,38)
    const float* f32 = (const float*)d_in[2];   // (16,255,19,19)
    float* out = (float*)d_out;                 // (16,100,6)

    // workspace carve-up (~8.3 MB total)
    float4* boxes  = (float4*)d_ws;
    float*  scores = (float*)((char*)d_ws + (size_t)NIMG * NPAD * sizeof(float4));
    int*    cls    = (int*)((char*)scores + (size_t)NIMG * NPAD * sizeof(float));

    const int T8  = NIMG * 3 * 76 * 76;   // 277248
    const int T16 = NIMG * 3 * 38 * 38;   //  69312
    const int T32 = NIMG * 3 * 19 * 19;   //  17328

    yolo_decode_kernel<76, 76><<<(T8 + 255) / 256, 256, 0, stream>>>(
        f8, boxes, scores, cls, 8.f,
        10.f, 13.f, 16.f, 30.f, 33.f, 23.f, OFF_S8);
    yolo_decode_kernel<38, 38><<<(T16 + 255) / 256, 256, 0, stream>>>(
        f16, boxes, scores, cls, 16.f,
        30.f, 61.f, 62.f, 45.f, 59.f, 119.f, OFF_S16);
    yolo_decode_kernel<19, 19><<<(T32 + 255) / 256, 256, 0, stream>>>(
        f32, boxes, scores, cls, 32.f,
        116.f, 90.f, 156.f, 198.f, 373.f, 326.f, OFF_S32);

    const size_t smem_bytes = (size_t)NPAD * sizeof(float)       // scores
                            + NMS_WAVES * sizeof(uint64_t)       // wave keys
                            + 8 * sizeof(float);                 // best bcast
    yolo_nms_kernel<<<NIMG, NMS_THREADS, smem_bytes, stream>>>(
        boxes, scores, cls, out);
}